// GudhiCubicalExtractor_51359218925741
// MI455X (gfx1250) — compile-verified
//
#include <hip/hip_runtime.h>
#include <cstdint>

#define NOFF   16129   // 127*127 relative offsets
#define NBUCKET 8192   // d2 max = 63^2+63^2 = 7938
#define NPIX   4096    // 64*64

typedef __attribute__((ext_vector_type(2))) float v2f;
typedef __attribute__((ext_vector_type(8))) float v8f;

#if __has_builtin(__builtin_amdgcn_wmma_f32_16x16x4_f32)
#define DTM_USE_WMMA 1
#endif

// ---------------------------------------------------------------------------
// Kernel 1: build the distance-sorted relative-offset table (deterministic).
// Entry = (d2 << 16) | (dy+63)<<8 | (dx+63), sorted ascending by d2.
// Counting sort: LDS histogram over integer d2, exclusive scan, then a
// deterministic rank (count of same-d2 offsets with smaller index, computed
// analytically via a perfect-square row scan) instead of atomic scatter.
// ---------------------------------------------------------------------------
__global__ __launch_bounds__(256) void dtm_build_table(unsigned int* __restrict__ table) {
  __shared__ int cnt[NBUCKET];
  __shared__ int chunk[256];
  const int tid = threadIdx.x;

  for (int i = tid; i < NBUCKET; i += 256) cnt[i] = 0;
  __syncthreads();

  for (int idx = tid; idx < NOFF; idx += 256) {
    int dy = idx / 127 - 63;
    int dx = idx % 127 - 63;
    atomicAdd(&cnt[dy * dy + dx * dx], 1);
  }
  __syncthreads();

  // exclusive scan: cnt[] (counts) -> cnt[] (bucket start offsets)
  const int base = tid * 32;
  int s = 0;
  for (int i = 0; i < 32; ++i) s += cnt[base + i];
  chunk[tid] = s;
  __syncthreads();
  if (tid == 0) {
    int run = 0;
    for (int i = 0; i < 256; ++i) { int v = chunk[i]; chunk[i] = run; run += v; }
  }
  __syncthreads();
  int run = chunk[tid];
  for (int i = 0; i < 32; ++i) { int v = cnt[base + i]; cnt[base + i] = run; run += v; }
  __syncthreads();

  // deterministic scatter
  for (int idx = tid; idx < NOFF; idx += 256) {
    int dy = idx / 127 - 63;
    int dx = idx % 127 - 63;
    int d2 = dy * dy + dx * dx;
    int rank = 0;
    // full rows ry < dy: offsets (ry, +-sq) with ry^2 + sq^2 == d2, |sq| <= 63
    for (int ry = -63; ry < dy; ++ry) {
      int r = d2 - ry * ry;
      if (r < 0) continue;
      int sq = (int)(sqrtf((float)r) + 0.5f);
      if (sq * sq != r) continue;
      if (sq == 0) rank += 1;
      else if (sq <= 63) rank += 2;
    }
    // same row: dx' = -dx precedes dx iff dx > 0
    if (dx > 0) rank += 1;
    int pos = cnt[d2] + rank;
    table[pos] = ((unsigned)d2 << 16) | ((unsigned)(dy + 63) << 8) | (unsigned)(dx + 63);
  }
}

// ---------------------------------------------------------------------------
// Kernel 2: DTM main. grid = (16 pixel-tiles, B batches), 256 threads (8 waves).
// Weights for the batch are DMA'd into LDS (async-to-LDS path), total mass is
// reduced with f32 WMMA (ones-vector matmul), then each thread walks the
// sorted offset table greedily for its pixel.
// ---------------------------------------------------------------------------
__global__ __launch_bounds__(256) void dtm_main(const float* __restrict__ x,
                                                const unsigned int* __restrict__ table,
                                                float* __restrict__ out) {
  __shared__ float sw[NPIX];     // 16 KB weight image for this batch
  __shared__ float red[256];
  const int tid = threadIdx.x;
  const int b = blockIdx.y;
  const float* src = x + (size_t)b * NPIX;

  // Async DMA global -> LDS: 4 passes x 256 lanes x 16 B = 16 KB
  for (int p = 0; p < 4; ++p) {
    int i = (p * 256 + tid) * 4;
    unsigned int ldsoff = (unsigned int)(uintptr_t)&sw[i];
    const float* g = src + i;
    asm volatile("global_load_async_to_lds_b128 %0, %1, off"
                 :: "v"(ldsoff), "v"(g) : "memory");
  }
  asm volatile("s_wait_asynccnt 0" ::: "memory");
  __syncthreads();

  float total;
#ifdef DTM_USE_WMMA
  {
    // Each wave reduces its 512-float chunk with 8 WMMAs:
    // D = A(16x4 weights) * B(4x16 ones) + C  => every D column = row-sums.
    const int wv = tid >> 5, lane = tid & 31;
    v8f c = {0.f, 0.f, 0.f, 0.f, 0.f, 0.f, 0.f, 0.f};
    v2f onesv; onesv.x = 1.0f; onesv.y = 1.0f;
    for (int k = 0; k < 8; ++k) {
      int bi = wv * 512 + k * 64 + lane * 2;
      v2f a; a.x = sw[bi]; a.y = sw[bi + 1];
      c = __builtin_amdgcn_wmma_f32_16x16x4_f32(false, a, false, onesv,
                                                (short)0, c, false, false);
    }
    float sacc = c[0] + c[1] + c[2] + c[3] + c[4] + c[5] + c[6] + c[7];
    sacc += __shfl_xor(sacc, 16, 32);   // lanes L and L+16 hold the two row halves
    if (lane == 0) red[wv] = sacc;
    __syncthreads();
    total = red[0] + red[1] + red[2] + red[3] + red[4] + red[5] + red[6] + red[7];
  }
#else
  {
    float p = 0.f;
    for (int i = tid; i < NPIX; i += 256) p += sw[i];
    red[tid] = p;
    __syncthreads();
    if (tid == 0) { float t = 0.f; for (int i = 0; i < 256; ++i) t += red[i]; red[0] = t; }
    __syncthreads();
    total = red[0];
  }
#endif

  const float mtarget = 0.01f * total;   // M0 = 0.01
  const int pix = blockIdx.x * 256 + tid;
  const int py = pix >> 6, px = pix & 63;

  float remaining = mtarget;
  float acc = 0.0f;
  for (int k = 0; k < NOFF; ++k) {
    if (!(remaining > 0.0f)) break;
    const unsigned e = table[k];
    const int ny = py + (int)((e >> 8) & 0xFFu) - 63;
    const int nx = px + (int)(e & 0xFFu) - 63;
    if ((unsigned)ny < 64u && (unsigned)nx < 64u) {
      const float w = sw[(ny << 6) + nx];
      const float sd = sqrtf((float)(e >> 16));   // match pow(sqrt(d2), 2.0)
      float take = fminf(w, remaining);
      take = fmaxf(take, 0.0f);
      acc = fmaf(take, sd * sd, acc);
      remaining -= take;
    }
  }

  float res = 0.0f;
  if (total > 0.0f) res = sqrtf(acc / mtarget);   // pow(x, 1/R) with R=2
  out[(size_t)b * NPIX + pix] = res;
}

// ---------------------------------------------------------------------------
extern "C" void kernel_launch(void* const* d_in, const int* in_sizes, int n_in,
                              void* d_out, int out_size, void* d_ws, size_t ws_size,
                              hipStream_t stream) {
  const float* x = (const float*)d_in[0];
  float* out = (float*)d_out;
  unsigned int* table = (unsigned int*)d_ws;   // 16129 * 4 B = 64.5 KB scratch
  const int B = in_sizes[0] / NPIX;            // 8 for the reference shapes

  dtm_build_table<<<dim3(1), dim3(256), 0, stream>>>(table);
  dtm_main<<<dim3(NPIX / 256, B), dim3(256), 0, stream>>>(x, table, out);
}